// ResidualSSM_56040733278828
// MI455X (gfx1250) — compile-verified
//
#include <hip/hip_runtime.h>
#include <hip/hip_bf16.h>
#include <math.h>
#include <stdint.h>

// ---------------- problem constants ----------------
#define BATCH    64
#define NWIN     512
#define DIN      1536
#define DSC      234
#define KIN      1770      // DIN + DSC
#define DM       64
#define NST      8
#define NCLS     234
#define MROWS    (BATCH * NWIN)   // 32768

typedef float v2f __attribute__((ext_vector_type(2)));
typedef float v8f __attribute__((ext_vector_type(8)));

// pointee type of the async-LDS builtins: 2 x i32 vector, AS1 (global) / AS3 (LDS)
typedef int v2i_t __attribute__((vector_size(8)));
typedef __attribute__((address_space(1))) v2i_t gv2i_t;
typedef __attribute__((address_space(3))) v2i_t lv2i_t;
typedef __attribute__((address_space(3))) void lds_void;

#if defined(__has_builtin)
#if __has_builtin(__builtin_amdgcn_global_load_async_to_lds_b64)
#define HAVE_ASYNC_BUILTIN 1
#endif
#if __has_builtin(__builtin_amdgcn_s_wait_asynccnt)
#define HAVE_WAIT_ASYNC_BUILTIN 1
#endif
#endif

static __device__ inline void async_copy8(const void* g, void* l) {
#if defined(HAVE_ASYNC_BUILTIN)
    __builtin_amdgcn_global_load_async_to_lds_b64(
        (gv2i_t*)(uintptr_t)g, (lv2i_t*)l, /*offset=*/0, /*cpol=*/0);
#else
    const unsigned loff = (unsigned)(unsigned long long)(lds_void*)l;
    asm volatile("global_load_async_to_lds_b64 %0, %1, off"
                 :: "v"(loff), "v"((unsigned long long)(uintptr_t)g)
                 : "memory");
#endif
}

#if defined(HAVE_WAIT_ASYNC_BUILTIN)
#define WAIT_ASYNC(n) __builtin_amdgcn_s_wait_asynccnt(n)
#else
#define WAIT_ASYNC(n) asm volatile("s_wait_asynccnt %0" :: "i"(n) : "memory")
#endif

static __device__ inline v8f wmma4(v2f a, v2f b, v8f c) {
    // D = A(16x4 f32) * B(4x16 f32) + C(16x16 f32)
    return __builtin_amdgcn_wmma_f32_16x16x4_f32(
        /*neg_a=*/false, a, /*neg_b=*/false, b,
        /*c_mod=*/(short)0, c, /*reuse_a=*/false, /*reuse_b=*/false);
}

static __device__ inline float gelu_exact(float x) {
    return 0.5f * x * (1.0f + erff(x * 0.70710678118654752440f));
}

static __device__ inline float softplus_f(float x) {
    return (x > 20.0f) ? x : log1pf(expf(x));
}

// ============================================================
// Kernel 1: meta vector  mv[b][d] = b_meta[d] + W_meta[d,:] . [site_emb; hour_emb]
// ============================================================
__global__ __launch_bounds__(64) void meta_kernel(
    const int* __restrict__ site_ids, const int* __restrict__ hours,
    const float* __restrict__ site_emb, const float* __restrict__ hour_emb,
    const float* __restrict__ W_meta, const float* __restrict__ b_meta,
    float* __restrict__ mv)
{
    const int b = blockIdx.x;
    const int d = threadIdx.x;
    int sid = site_ids[b]; sid = sid < 0 ? 0 : (sid > 19 ? 19 : sid);
    int hr  = hours[b];    hr  = hr  < 0 ? 0 : (hr  > 23 ? 23 : hr);
    const float* wr = W_meta + d * 16;
    float acc = b_meta[d];
    #pragma unroll
    for (int k = 0; k < 8; ++k) acc += wr[k]     * site_emb[sid * 8 + k];
    #pragma unroll
    for (int k = 0; k < 8; ++k) acc += wr[8 + k] * hour_emb[hr * 8 + k];
    mv[b * 64 + d] = acc;
}

// ============================================================
// Kernel 2: input projection GEMM (M=32768, N=64, K=1770) via WMMA
//   async global->LDS double-buffered staging, fused LN1/GELU/pos/meta epilogue
// block = 128 threads (4 waves); 16 rows per block, wave w owns n-tile 16w..
// ============================================================

// issue one 128-wide K chunk (fully valid: chunks 0..12) as async b64 copies.
// 8 (A) + 32 (B) = 40 uniform async instructions per wave.
static __device__ inline void issue_chunk(
    const float* __restrict__ emb, const float* __restrict__ fp,
    const float* __restrict__ W_in, int r0, int k0,
    float (*A)[132], float (*B)[132], int t)
{
    // A tile: 16 rows x 128 floats, 8-byte segments (fp rows are only 8B-aligned)
    #pragma unroll
    for (int j = 0; j < 8; ++j) {
        const int seg = t + j * 128;        // 0..1023
        const int r   = seg >> 6;           // 64 segments per row
        const int kk  = (seg & 63) * 2;
        const int k   = k0 + kk;
        const float* g = (k < DIN)
            ? emb + (size_t)(r0 + r) * DIN + k
            : fp  + (size_t)(r0 + r) * DSC + (k - DIN);
        async_copy8(g, &A[r][kk]);
    }
    // B tile: 64 rows x 128 floats (W_in rows are only 8B-aligned)
    #pragma unroll
    for (int j = 0; j < 32; ++j) {
        const int seg = t + j * 128;        // 0..4095
        const int r   = seg >> 6;
        const int kk  = (seg & 63) * 2;
        async_copy8(W_in + (size_t)r * KIN + (k0 + kk), &B[r][kk]);
    }
}

static __device__ inline void mm_chunk(
    const float (*A)[132], const float (*B)[132],
    int ln, int half, int n0, v8f& acc0, v8f& acc1)
{
    #pragma unroll
    for (int kk = 0; kk < 128; kk += 8) {
        v2f a0 = *(const v2f*)&A[ln][kk + 2 * half];
        v2f b0 = *(const v2f*)&B[n0 + ln][kk + 2 * half];
        acc0 = wmma4(a0, b0, acc0);
        v2f a1 = *(const v2f*)&A[ln][kk + 4 + 2 * half];
        v2f b1 = *(const v2f*)&B[n0 + ln][kk + 4 + 2 * half];
        acc1 = wmma4(a1, b1, acc1);
    }
}

__global__ __launch_bounds__(128) void in_proj_kernel(
    const float* __restrict__ emb, const float* __restrict__ fp,
    const float* __restrict__ W_in, const float* __restrict__ b_in,
    const float* __restrict__ g1, const float* __restrict__ beta1,
    const float* __restrict__ pos, const float* __restrict__ mv,
    float* __restrict__ hbuf)
{
    __shared__ float Axs[2][16][132];   // double-buffered A
    __shared__ float Bxs[2][64][132];   // double-buffered W_in
    __shared__ float zt[16][66];
    __shared__ float mu[16], rsd[16];

    const int t    = threadIdx.x;
    const int lane = t & 31;
    const int wid  = t >> 5;
    const int half = lane >> 4;
    const int ln   = lane & 15;
    const int r0   = blockIdx.x * 16;
    const int n0   = wid * 16;

    v8f acc0 = {0.f, 0.f, 0.f, 0.f, 0.f, 0.f, 0.f, 0.f};
    v8f acc1 = {0.f, 0.f, 0.f, 0.f, 0.f, 0.f, 0.f, 0.f};

    // ---- async double-buffered pipeline over chunks 0..12 ----
    issue_chunk(emb, fp, W_in, r0, 0, Axs[0], Bxs[0], t);
    int cur = 0;
    for (int c = 0; c < 13; ++c) {
        if (c < 12) {
            issue_chunk(emb, fp, W_in, r0, (c + 1) * 128,
                        Axs[cur ^ 1], Bxs[cur ^ 1], t);
            WAIT_ASYNC(40);   // <=40 outstanding => chunk c has landed
        } else {
            WAIT_ASYNC(0);
        }
        __syncthreads();
        mm_chunk(Axs[cur], Bxs[cur], ln, half, n0, acc0, acc1);
        __syncthreads();
        cur ^= 1;
    }

    // ---- tail chunk 13 (k0 = 1664): scalar staging with zero pad ----
    {
        const int k0 = 13 * 128;
        for (int idx = t; idx < 16 * 128; idx += 128) {
            const int r = idx >> 7, kk = idx & 127, k = k0 + kk;
            float v = 0.f;
            if (k < KIN) v = fp[(size_t)(r0 + r) * DSC + (k - DIN)];
            Axs[cur][r][kk] = v;
        }
        for (int idx = t; idx < 64 * 128; idx += 128) {
            const int n = idx >> 7, kk = idx & 127, k = k0 + kk;
            Bxs[cur][n][kk] = (k < KIN) ? W_in[(size_t)n * KIN + k] : 0.f;
        }
        __syncthreads();
        mm_chunk(Axs[cur], Bxs[cur], ln, half, n0, acc0, acc1);
        __syncthreads();
    }

    const v8f acc = acc0 + acc1;

    // write tile to LDS with bias
    #pragma unroll
    for (int v = 0; v < 8; ++v) {
        const int r = v + 8 * half;
        const int c = n0 + ln;
        zt[r][c] = acc[v] + b_in[c];
    }
    __syncthreads();

    // per-row LayerNorm stats
    if (t < 16) {
        float s = 0.f;
        for (int c = 0; c < 64; ++c) s += zt[t][c];
        const float m_ = s * (1.0f / 64.0f);
        float s2 = 0.f;
        for (int c = 0; c < 64; ++c) { float d = zt[t][c] - m_; s2 += d * d; }
        mu[t]  = m_;
        rsd[t] = rsqrtf(s2 * (1.0f / 64.0f) + 1e-5f);
    }
    __syncthreads();

    // LN -> GELU -> +pos +meta -> hbuf
    for (int i = t; i < 16 * 64; i += 128) {
        const int r = i >> 6;
        const int c = i & 63;
        const int m = r0 + r;
        float x = (zt[r][c] - mu[r]) * rsd[r] * g1[c] + beta1[c];
        x = gelu_exact(x);
        const int bi = m >> 9;     // batch
        const int wi = m & 511;    // window
        x += pos[wi * 64 + c] + mv[bi * 64 + c];
        hbuf[(size_t)m * 64 + c] = x;
    }
}

// ============================================================
// Kernel 3: per-timestep SSM projections, both directions, one GEMM
//   P[m, 0:64]    = softplus(h @ fwd_Wdt.T + fwd_bdt)
//   P[m, 64:80]   = h @ fwd_Wx.T + fwd_bx      (B then C)
//   P[m, 80:144]  = softplus(h @ bwd_Wdt.T + bwd_bdt)
//   P[m, 144:160] = h @ bwd_Wx.T + bwd_bx
// M=32768, N=160, K=64
// ============================================================
__global__ __launch_bounds__(128) void proj_kernel(
    const float* __restrict__ hbuf,
    const float* __restrict__ fWdt, const float* __restrict__ fbdt,
    const float* __restrict__ fWx,  const float* __restrict__ fbx,
    const float* __restrict__ bWdt, const float* __restrict__ bbdt,
    const float* __restrict__ bWx,  const float* __restrict__ bbx,
    float* __restrict__ P)
{
    __shared__ float As[16][66];
    __shared__ float Bs[160][66];

    const int t    = threadIdx.x;
    const int lane = t & 31;
    const int wid  = t >> 5;
    const int half = lane >> 4;
    const int ln   = lane & 15;
    const int r0   = blockIdx.x * 16;

    for (int idx = t; idx < 16 * 64; idx += 128)
        As[idx >> 6][idx & 63] = hbuf[(size_t)(r0 + (idx >> 6)) * 64 + (idx & 63)];

    for (int idx = t; idx < 160 * 64; idx += 128) {
        const int n = idx >> 6;
        const int k = idx & 63;
        const float* wr;
        if (n < 64)       wr = fWdt + (size_t)n * 64;
        else if (n < 80)  wr = fWx  + (size_t)(n - 64) * 64;
        else if (n < 144) wr = bWdt + (size_t)(n - 80) * 64;
        else              wr = bWx  + (size_t)(n - 144) * 64;
        Bs[n][k] = wr[k];
    }
    __syncthreads();

    for (int ti = wid; ti < 10; ti += 4) {
        const int n0 = ti * 16;
        v8f acc = {0.f, 0.f, 0.f, 0.f, 0.f, 0.f, 0.f, 0.f};
        #pragma unroll
        for (int kk = 0; kk < 64; kk += 4) {
            v2f a = *(const v2f*)&As[ln][kk + 2 * half];
            v2f b = *(const v2f*)&Bs[n0 + ln][kk + 2 * half];
            acc = wmma4(a, b, acc);
        }
        #pragma unroll
        for (int v = 0; v < 8; ++v) {
            const int r = v + 8 * half;
            const int c = n0 + ln;
            float bias; bool sp;
            if (c < 64)       { bias = fbdt[c];        sp = true;  }
            else if (c < 80)  { bias = fbx[c - 64];    sp = false; }
            else if (c < 144) { bias = bbdt[c - 80];   sp = true;  }
            else              { bias = bbx[c - 144];   sp = false; }
            float val = acc[v] + bias;
            if (sp) val = softplus_f(val);
            P[(size_t)(r0 + r) * 160 + c] = val;
        }
    }
}

// ============================================================
// Kernel 4: selective scan. grid = (batch, dir), block = 256 threads.
// Thread t owns states (d = t&63, n = t>>6) and (d, n+4) in registers.
// Writes y into merged[m*128 + dir*64 + d].
// ============================================================
__global__ __launch_bounds__(256) void scan_kernel(
    const float* __restrict__ hbuf, const float* __restrict__ P,
    const float* __restrict__ fAlog, const float* __restrict__ fD,
    const float* __restrict__ bAlog, const float* __restrict__ bD,
    float* __restrict__ merged)
{
    __shared__ float xs[64], dls[64], bcs[16], part[256];

    const int t   = threadIdx.x;
    const int b   = blockIdx.x;
    const int dir = blockIdx.y;

    const float* Alog = dir ? bAlog : fAlog;
    const float* Dp   = dir ? bD    : fD;
    const int doff = dir ? 80  : 0;
    const int boff = dir ? 144 : 64;   // B: boff..boff+7, C: boff+8..boff+15

    const int d = t & 63;
    const int g = t >> 6;              // 0..3

    const float A0 = -expf(Alog[d * NST + g]);
    const float A1 = -expf(Alog[d * NST + g + 4]);
    float h0 = 0.f, h1 = 0.f;

    for (int s = 0; s < NWIN; ++s) {
        const int tt = dir ? (NWIN - 1 - s) : s;
        const size_t m = (size_t)b * NWIN + tt;

        if (t < 64) {
            xs[t]  = hbuf[m * 64 + t];
            dls[t] = P[m * 160 + doff + t];
        } else if (t < 80) {
            bcs[t - 64] = P[m * 160 + boff + (t - 64)];
        }
        __syncthreads();

        const float delta = dls[d];
        const float xv    = xs[d];
        h0 = expf(delta * A0) * h0 + delta * bcs[g]     * xv;
        h1 = expf(delta * A1) * h1 + delta * bcs[g + 4] * xv;
        part[t] = h0 * bcs[8 + g] + h1 * bcs[8 + g + 4];
        __syncthreads();

        if (t < 64) {
            const float y = part[t] + part[64 + t] + part[128 + t] + part[192 + t]
                          + Dp[t] * xs[t];
            merged[m * 128 + dir * 64 + t] = y;
        }
        __syncthreads();
    }
}

// ============================================================
// Kernel 5: merge GEMM (K=128,N=64) + residual + LN2 + head GEMM (K=64,N=234)
// block = 128 threads (4 waves), 16 rows per block
// ============================================================
__global__ __launch_bounds__(128) void merge_head_kernel(
    const float* __restrict__ merged, const float* __restrict__ hbuf,
    const float* __restrict__ Wm, const float* __restrict__ bm,
    const float* __restrict__ g2, const float* __restrict__ beta2,
    const float* __restrict__ Wh, const float* __restrict__ bh,
    float* __restrict__ out)
{
    __shared__ float Ams[16][130];
    __shared__ float Wms[64][130];
    __shared__ float zt[16][66];
    __shared__ float zln[16][66];
    __shared__ float mu[16], rsd[16];

    const int t    = threadIdx.x;
    const int lane = t & 31;
    const int wid  = t >> 5;
    const int half = lane >> 4;
    const int ln   = lane & 15;
    const int r0   = blockIdx.x * 16;
    const int n0   = wid * 16;

    for (int idx = t; idx < 16 * 128; idx += 128)
        Ams[idx >> 7][idx & 127] = merged[(size_t)(r0 + (idx >> 7)) * 128 + (idx & 127)];
    for (int idx = t; idx < 64 * 128; idx += 128)
        Wms[idx >> 7][idx & 127] = Wm[(size_t)(idx >> 7) * 128 + (idx & 127)];
    __syncthreads();

    // merge GEMM
    v8f acc = {0.f, 0.f, 0.f, 0.f, 0.f, 0.f, 0.f, 0.f};
    #pragma unroll
    for (int kk = 0; kk < 128; kk += 4) {
        v2f a = *(const v2f*)&Ams[ln][kk + 2 * half];
        v2f b = *(const v2f*)&Wms[n0 + ln][kk + 2 * half];
        acc = wmma4(a, b, acc);
    }
    #pragma unroll
    for (int v = 0; v < 8; ++v) {
        const int r = v + 8 * half;
        const int c = n0 + ln;
        zt[r][c] = acc[v] + bm[c] + hbuf[(size_t)(r0 + r) * 64 + c];  // + residual
    }
    __syncthreads();

    if (t < 16) {
        float s = 0.f;
        for (int c = 0; c < 64; ++c) s += zt[t][c];
        const float m_ = s * (1.0f / 64.0f);
        float s2 = 0.f;
        for (int c = 0; c < 64; ++c) { float dd = zt[t][c] - m_; s2 += dd * dd; }
        mu[t]  = m_;
        rsd[t] = rsqrtf(s2 * (1.0f / 64.0f) + 1e-5f);
    }
    __syncthreads();

    for (int i = t; i < 16 * 64; i += 128) {
        const int r = i >> 6;
        const int c = i & 63;
        zln[r][c] = (zt[r][c] - mu[r]) * rsd[r] * g2[c] + beta2[c];
    }
    __syncthreads();

    // head GEMM: 15 n-tiles (240 padded, valid n < 234)
    for (int ti = wid; ti < 15; ti += 4) {
        const int nh = ti * 16;
        v8f acc2 = {0.f, 0.f, 0.f, 0.f, 0.f, 0.f, 0.f, 0.f};
        const int n = nh + ln;
        #pragma unroll
        for (int kk = 0; kk < 64; kk += 4) {
            v2f a = *(const v2f*)&zln[ln][kk + 2 * half];
            v2f b = {0.f, 0.f};
            if (n < NCLS) b = *(const v2f*)&Wh[(size_t)n * 64 + kk + 2 * half];
            acc2 = wmma4(a, b, acc2);
        }
        #pragma unroll
        for (int v = 0; v < 8; ++v) {
            const int r = v + 8 * half;
            const int c = nh + ln;
            if (c < NCLS)
                out[(size_t)(r0 + r) * NCLS + c] = acc2[v] + bh[c];
        }
    }
}

// ============================================================
// host launcher
// ============================================================
extern "C" void kernel_launch(void* const* d_in, const int* in_sizes, int n_in,
                              void* d_out, int out_size, void* d_ws, size_t ws_size,
                              hipStream_t stream) {
    (void)in_sizes; (void)n_in; (void)out_size; (void)ws_size;

    // setup_inputs() dict order
    const float* emb      = (const float*)d_in[0];
    const float* fp       = (const float*)d_in[1];
    const int*   site_ids = (const int*)  d_in[2];
    const int*   hours    = (const int*)  d_in[3];
    const float* W_in     = (const float*)d_in[4];
    const float* b_in     = (const float*)d_in[5];
    const float* ln1_g    = (const float*)d_in[6];
    const float* ln1_b    = (const float*)d_in[7];
    const float* site_emb = (const float*)d_in[8];
    const float* hour_emb = (const float*)d_in[9];
    const float* W_meta   = (const float*)d_in[10];
    const float* b_meta   = (const float*)d_in[11];
    const float* pos_enc  = (const float*)d_in[12];
    const float* W_merge  = (const float*)d_in[13];
    const float* b_merge  = (const float*)d_in[14];
    const float* ln2_g    = (const float*)d_in[15];
    const float* ln2_b    = (const float*)d_in[16];
    const float* W_head   = (const float*)d_in[17];
    const float* b_head   = (const float*)d_in[18];
    const float* fAlog    = (const float*)d_in[19];
    const float* fD       = (const float*)d_in[20];
    const float* fWx      = (const float*)d_in[21];
    const float* fbx      = (const float*)d_in[22];
    const float* fWdt     = (const float*)d_in[23];
    const float* fbdt     = (const float*)d_in[24];
    const float* bAlog    = (const float*)d_in[25];
    const float* bD       = (const float*)d_in[26];
    const float* bWx      = (const float*)d_in[27];
    const float* bbx      = (const float*)d_in[28];
    const float* bWdt     = (const float*)d_in[29];
    const float* bbdt     = (const float*)d_in[30];

    // workspace layout
    char* w = (char*)d_ws;
    const size_t H_BYTES  = (size_t)MROWS * 64  * sizeof(float);  // 8 MB
    const size_t MV_BYTES = (size_t)BATCH * 64  * sizeof(float);  // 16 KB
    const size_t P_BYTES  = (size_t)MROWS * 160 * sizeof(float);  // 20 MB
    float* hbuf   = (float*)(w);
    float* mv     = (float*)(w + H_BYTES);
    float* P      = (float*)(w + H_BYTES + MV_BYTES);
    float* merged = (float*)(w + H_BYTES + MV_BYTES + P_BYTES);   // 16 MB

    float* out = (float*)d_out;

    meta_kernel<<<BATCH, 64, 0, stream>>>(site_ids, hours, site_emb, hour_emb,
                                          W_meta, b_meta, mv);

    in_proj_kernel<<<MROWS / 16, 128, 0, stream>>>(emb, fp, W_in, b_in,
                                                   ln1_g, ln1_b, pos_enc, mv, hbuf);

    proj_kernel<<<MROWS / 16, 128, 0, stream>>>(hbuf, fWdt, fbdt, fWx, fbx,
                                                bWdt, bbdt, bWx, bbx, P);

    scan_kernel<<<dim3(BATCH, 2), 256, 0, stream>>>(hbuf, P, fAlog, fD,
                                                    bAlog, bD, merged);

    merge_head_kernel<<<MROWS / 16, 128, 0, stream>>>(merged, hbuf, W_merge, b_merge,
                                                      ln2_g, ln2_b, W_head, b_head, out);
}